// MessagePassing_28346784154210
// MI455X (gfx1250) — compile-verified
//
#include <hip/hip_runtime.h>

// R-GCN basis-decomposition message passing for MI455X (gfx1250, wave32).
//
// msg_e = sum_b comps[r_e,b] * (x[col_e] @ bases[b]) * alpha[r_e] * norm_e
// out[row_e] += msg_e
//
// The 4 shared 128x128 f32 bases (256 KB) live in LDS (320 KB/WGP).
// Each wave processes a 16-edge tile: gathers a 16x128 A tile of source rows,
// then runs 32(K-steps) x 4(bases) V_WMMA_F32_16X16X4_F32 per 16-col N-tile
// into FOUR independent accumulators (one per base): interleaved C-chains hide
// WMMA accumulation latency and the hot loop is pure ds_load + v_wmma.
// LDS layout is [K-pair][base][n] so one K-step's four B fragments sit at
// small immediate DS offsets ((8*kk+b)*1024 B) from a single lane base.
// The per-edge scalar coef_{e,b} = comps[r,b]*alpha[r]*norm_e is applied in
// the epilogue via per-row shuffled coefficients, then scattered with
// global_atomic_add_f32.

#define DDIM  128
#define NBASE 4

typedef __attribute__((ext_vector_type(2))) float v2f;
typedef __attribute__((ext_vector_type(8))) float v8f;

__global__ void rgcn_count_deg(const long long* __restrict__ eidx,
                               float* __restrict__ deg, int E) {
  int e = blockIdx.x * blockDim.x + threadIdx.x;
  if (e < E) unsafeAtomicAdd(deg + (int)eidx[e], 1.0f);
}

__global__ void rgcn_deg_inv(float* __restrict__ deg, int n) {
  int i = blockIdx.x * blockDim.x + threadIdx.x;
  if (i < n) {
    float d = deg[i];
    deg[i] = (d > 0.0f) ? rsqrtf(fmaxf(d, 1.0f)) : 0.0f;
  }
}

// Interleave bases for single-ds_load_b64 B fragments, [K-pair][base][n]:
// BL[(kk2*NBASE + b)*128 + n] = ( bases[b][2*kk2][n], bases[b][2*kk2+1][n] ).
__global__ void rgcn_prep_bases(const float* __restrict__ bases,
                                float2* __restrict__ BL) {
  int i = blockIdx.x * blockDim.x + threadIdx.x;
  if (i < 64 * NBASE * DDIM) {
    int n   = i & (DDIM - 1);
    int b   = (i >> 7) & (NBASE - 1);
    int kk2 = i >> 9;
    const float* base = bases + (size_t)b * DDIM * DDIM;
    BL[i] = make_float2(base[(2 * kk2) * DDIM + n],
                        base[(2 * kk2 + 1) * DDIM + n]);
  }
}

__launch_bounds__(256)
__global__ void rgcn_wmma_main(const float* __restrict__ x,
                               const float* __restrict__ comps,
                               const float* __restrict__ alpha,
                               const long long* __restrict__ eidx,
                               const long long* __restrict__ etype,
                               const float* __restrict__ deginv,
                               const float2* __restrict__ BLg,
                               float* __restrict__ out,
                               int E, int nTiles) {
  extern __shared__ float2 smem[];  // 64*NBASE*DDIM float2 = 256 KB

  // Cooperative stage: all 4 interleaved bases -> LDS (amortized over tiles).
  {
    const float4* src = (const float4*)BLg;
    float4* dst = (float4*)smem;
    const int n4 = 64 * NBASE * DDIM / 2;  // 16384 float4
    for (int i = threadIdx.x; i < n4; i += blockDim.x) dst[i] = src[i];
  }
  __syncthreads();

  const int lane = threadIdx.x & 31;
  const int m    = lane & 15;     // A-matrix row owned by this lane
  const int hi   = lane >> 4;     // half-wave: K pair (0,1) vs (2,3)
  const int wave = threadIdx.x >> 5;
  const int wavesPerBlk = blockDim.x >> 5;
  const int gwave  = blockIdx.x * wavesPerBlk + wave;
  const int nwaves = gridDim.x * wavesPerBlk;

  for (int t = gwave; t < nTiles; t += nwaves) {
    const int e0 = t << 4;
    const int e  = e0 + m;
    const int ec = (e < E) ? e : (E - 1);

    const int rown = (int)eidx[ec];
    const int coln = (int)eidx[(size_t)E + ec];
    const int rel  = (int)etype[ec];

    float sc = deginv[rown] * deginv[coln] * alpha[rel];
    if (e >= E) sc = 0.0f;  // tail tile: zero contribution
    const float4 cv = *(const float4*)(comps + rel * NBASE);
    const float c0 = cv.x * sc, c1 = cv.y * sc, c2 = cv.z * sc, c3 = cv.w * sc;

    // Gather this lane's (unscaled) A fragments: row m of the 16x128 tile.
    // 16x4 f32 A layout: lanes 0-15 hold (K0,K1), lanes 16-31 hold (K2,K3).
    const float* xrow = x + (size_t)coln * DDIM;
    v2f a[32];
#pragma unroll
    for (int kk = 0; kk < 32; ++kk) {
      const float2 v = *(const float2*)(xrow + 4 * kk + 2 * hi);
      a[kk] = (v2f){v.x, v.y};
    }

    // Per-output-row data for the epilogue: C/D VGPR i holds row (i + 8*hi).
    int   rs[8];
    float cr0[8], cr1[8], cr2[8], cr3[8];
#pragma unroll
    for (int i = 0; i < 8; ++i) {
      const int src = i + 8 * hi;
      rs[i]  = __shfl(rown, src, 32);
      cr0[i] = __shfl(c0, src, 32);
      cr1[i] = __shfl(c1, src, 32);
      cr2[i] = __shfl(c2, src, 32);
      cr3[i] = __shfl(c3, src, 32);
    }

    for (int nt = 0; nt < 8; ++nt) {
      const int n0 = nt << 4;
      // Lane's B fragment base: column n0+m; half-wave picks the odd K-pair
      // ((K2,K3)) at +NBASE*DDIM. Fragment (kk,b) is then a small immediate
      // offset: ((2*kk)*NBASE + b)*DDIM float2 = (8*kk + b)*1024 bytes.
      const float2* bl = smem + (n0 + m) + (size_t)hi * (NBASE * DDIM);
      v8f acc0 = (v8f){0.f, 0.f, 0.f, 0.f, 0.f, 0.f, 0.f, 0.f};
      v8f acc1 = acc0, acc2 = acc0, acc3 = acc0;
#pragma unroll
      for (int kk = 0; kk < 32; ++kk) {
        const float2 b0 = bl[((2 * kk) * NBASE + 0) * DDIM];
        const float2 b1 = bl[((2 * kk) * NBASE + 1) * DDIM];
        const float2 b2 = bl[((2 * kk) * NBASE + 2) * DDIM];
        const float2 b3 = bl[((2 * kk) * NBASE + 3) * DDIM];
        // 4 independent accumulator chains: no VALU in the hot loop,
        // distance-4 between same-chain WMMAs hides C-RAW latency.
        acc0 = __builtin_amdgcn_wmma_f32_16x16x4_f32(
            false, a[kk], false, (v2f){b0.x, b0.y}, (short)0, acc0, false, false);
        acc1 = __builtin_amdgcn_wmma_f32_16x16x4_f32(
            false, a[kk], false, (v2f){b1.x, b1.y}, (short)0, acc1, false, false);
        acc2 = __builtin_amdgcn_wmma_f32_16x16x4_f32(
            false, a[kk], false, (v2f){b2.x, b2.y}, (short)0, acc2, false, false);
        acc3 = __builtin_amdgcn_wmma_f32_16x16x4_f32(
            false, a[kk], false, (v2f){b3.x, b3.y}, (short)0, acc3, false, false);
      }
      // Epilogue: combine bases with per-row coefs, scatter 16x16 f32 tile.
      // acc*[i] is row (i + 8*hi), column n0+m.
      float* outc = out + n0 + m;
#pragma unroll
      for (int i = 0; i < 8; ++i) {
        const float val = cr0[i] * acc0[i] + cr1[i] * acc1[i] +
                          cr2[i] * acc2[i] + cr3[i] * acc3[i];
        if (e0 + i + 8 * hi < E)
          unsafeAtomicAdd(outc + (size_t)rs[i] * DDIM, val);
      }
    }
  }
}

extern "C" void kernel_launch(void* const* d_in, const int* in_sizes, int n_in,
                              void* d_out, int out_size, void* d_ws, size_t ws_size,
                              hipStream_t stream) {
  const float*     x     = (const float*)d_in[0];
  const float*     bases = (const float*)d_in[1];
  const float*     comps = (const float*)d_in[2];
  const float*     alpha = (const float*)d_in[3];
  const long long* eidx  = (const long long*)d_in[4];
  const long long* etype = (const long long*)d_in[5];

  const int nent   = in_sizes[0] / DDIM;
  const int E      = in_sizes[5];
  const int nTiles = (E + 15) / 16;

  // Workspace: [deg/deg_inv: nent floats][BL: 64*4*128 float2 = 256 KB]
  float*  deg = (float*)d_ws;
  float2* BL  = (float2*)((char*)d_ws + (size_t)nent * sizeof(float));

  hipMemsetAsync(d_out, 0, (size_t)out_size * sizeof(float), stream);
  hipMemsetAsync(deg, 0, (size_t)nent * sizeof(float), stream);

  rgcn_count_deg<<<(E + 255) / 256, 256, 0, stream>>>(eidx, deg, E);
  rgcn_deg_inv<<<(nent + 255) / 256, 256, 0, stream>>>(deg, nent);
  rgcn_prep_bases<<<(64 * NBASE * DDIM + 255) / 256, 256, 0, stream>>>(bases, BL);

  const size_t shbytes = (size_t)64 * NBASE * DDIM * sizeof(float2);  // 256 KB
  hipFuncSetAttribute((const void*)rgcn_wmma_main,
                      hipFuncAttributeMaxDynamicSharedMemorySize, (int)shbytes);

  int blocks = 1024;  // grid-stride; amortizes the 256 KB LDS staging
  if (blocks * 8 > nTiles) blocks = (nTiles + 7) / 8;
  if (blocks < 1) blocks = 1;
  rgcn_wmma_main<<<blocks, 256, shbytes, stream>>>(
      x, comps, alpha, eidx, etype, deg, BL, (float*)d_out, E, nTiles);
}